// SpikeFP32Embedding_11450382811508
// MI455X (gfx1250) — compile-verified
//
#include <hip/hip_runtime.h>
#include <stdint.h>

// SpikeFP32Embedding: per-token row gather from the bit-pulse table.
//   token_ids    : int32  [B*S]             (B=8, S=2048 -> 16384 tokens)
//   weight_pulse : float32 [32768, 128, 32] (16 KB per row)
//   out          : float32 [B*S, 128, 32]
//
// Zero-FLOP gather/copy: ~256 MB streamed out + ~200 MB unique table reads
// (mostly L2-resident; L2 = 192 MB). Floor ~20 us @ 23.3 TB/s.
// CDNA5 async global<->LDS DMA path (ASYNCcnt), chunk strides folded into
// the instruction's 24-bit IOFFSET (applied to both LDS and global address
// per the ISA pseudocode), NT output stores to protect L2.

#define ROW_FLOATS 4096          // 128 * 32 floats per token row (16 KB)
#define THREADS    256           // 8 waves (wave32)

__global__ __launch_bounds__(THREADS)
void spike_embed_gather_kernel(const int* __restrict__ token_ids,
                               const float* __restrict__ weight_pulse,
                               float* __restrict__ out)
{
    __shared__ float sh[ROW_FLOATS];          // 16 KB staging buffer

    const int tok = token_ids[blockIdx.x];    // uniform per block -> SGPR

    // Per-lane base addresses; chunk strides go in the offset: immediate.
    const uint32_t lane16 = threadIdx.x * 16u;            // 16 B per lane
    const uint64_t src = (uint64_t)(uintptr_t)weight_pulse +
                         (uint64_t)(unsigned)tok * (ROW_FLOATS * 4u) + lane16;
    const uint64_t dst = (uint64_t)(uintptr_t)out +
                         (uint64_t)blockIdx.x * (ROW_FLOATS * 4u) + lane16;

    // Low 32 bits of the flat pointer into LDS == LDS byte offset
    // (LDS aperture mapping).
    const uint32_t lds = (uint32_t)(uintptr_t)(void*)sh + lane16;

    // Async DMA global -> LDS: 16 B/lane, 512 B/wave per issue.
    // IOFFSET is added to BOTH the LDS and the global address, so one base
    // address pair serves all four 4 KB chunks (no per-chunk VALU math).
    asm volatile(
        "global_load_async_to_lds_b128 %0, %1, off\n\t"
        "global_load_async_to_lds_b128 %0, %1, off offset:4096\n\t"
        "global_load_async_to_lds_b128 %0, %1, off offset:8192\n\t"
        "global_load_async_to_lds_b128 %0, %1, off offset:12288"
        :
        : "v"(lds), "v"(src)
        : "memory");

    // Wait for this wave's async loads to land in LDS. Each wave loads and
    // stores only its own lanes' LDS slice -> no workgroup barrier needed.
    asm volatile("s_wait_asynccnt 0" ::: "memory");

    // Async DMA LDS -> global, non-temporal (output is write-once; keep L2
    // capacity for the reused embedding rows).
    asm volatile(
        "global_store_async_from_lds_b128 %0, %1, off th:TH_STORE_NT\n\t"
        "global_store_async_from_lds_b128 %0, %1, off offset:4096 th:TH_STORE_NT\n\t"
        "global_store_async_from_lds_b128 %0, %1, off offset:8192 th:TH_STORE_NT\n\t"
        "global_store_async_from_lds_b128 %0, %1, off offset:12288 th:TH_STORE_NT"
        :
        : "v"(dst), "v"(lds)
        : "memory");
    // s_endpgm performs an implicit wait-idle, draining ASYNCcnt so the
    // async stores are globally visible when the kernel retires.
}

extern "C" void kernel_launch(void* const* d_in, const int* in_sizes, int n_in,
                              void* d_out, int out_size, void* d_ws, size_t ws_size,
                              hipStream_t stream) {
    (void)n_in; (void)out_size; (void)d_ws; (void)ws_size;

    const int*   token_ids    = (const int*)d_in[0];     // [16384]
    const float* weight_pulse = (const float*)d_in[1];   // [32768*128*32]
    float*       out          = (float*)d_out;           // [16384*128*32]

    const int n_tokens = in_sizes[0];                    // 16384 blocks
    spike_embed_gather_kernel<<<n_tokens, THREADS, 0, stream>>>(
        token_ids, weight_pulse, out);
}